// J_crf_23450521437029
// MI455X (gfx1250) — compile-verified
//
#include <hip/hip_runtime.h>
#include <stdint.h>

// CRF NLL for B=2048, T=4096, K=3 on MI455X (gfx1250, wave32).
// Memory-bound: ~150-190 MB of traffic -> ~7-8us at 23.3 TB/s.
// Parallel scan over T via the (log,+,logsumexp) semiring; chunk matrices are
// built in the exp domain with power-of-two rescaling (3 transcendentals/step).
// CDNA5-specific: global_load_async_to_lds_b128 staging + s_wait_asynccnt.

#define B_SZ   2048
#define T_SZ   4096
#define K_TAGS 3
#define L_CH   32
#define N_CH   (T_SZ / L_CH)   // 128 chunks per batch row

#define L2E 1.4426950408889634f
#define LN2 0.6931471805599453f

__device__ __forceinline__ float lse3(float a, float b, float c) {
    float m = fmaxf(a, fmaxf(b, c));
    if (m < -1e30f) return m;  // guard: all terms ~ -inf
    float s = __builtin_exp2f((a - m) * L2E)
            + __builtin_exp2f((b - m) * L2E)
            + __builtin_exp2f((c - m) * L2E);
    return m + __builtin_log2f(s) * LN2;
}

// ---------------------------------------------------------------------------
// Phase 1: one block per batch row b. Async-stage the 48KB logits row into
// LDS, then each thread builds the 3x3 semiring matrix of its 32-step chunk
// (exp domain, 2^e rescale) and accumulates emit/trans score partials.
// ---------------------------------------------------------------------------
__launch_bounds__(N_CH)
__global__ void crf_phase1(const float* __restrict__ logits,
                           const float* __restrict__ trans,
                           const int*   __restrict__ tags,
                           float* __restrict__ chunkM,   // [B*N_CH*9]
                           float* __restrict__ scoreOut) // [B]
{
    __shared__ float sTrans[16];
    __shared__ float sLog[T_SZ * K_TAGS];   // 48 KB
    __shared__ float sRed[N_CH];

    const int b   = blockIdx.x;
    const int tid = threadIdx.x;

    if (tid < 9) sTrans[tid] = trans[tid];

    // --- Async global -> LDS staging: 3072 x 16B, coalesced 512B/wave/instr.
    const float* gRow = logits + (size_t)b * (T_SZ * K_TAGS);
    {
        const unsigned ldsBase = (unsigned)(uintptr_t)&sLog[0];
#pragma unroll
        for (int it = 0; it < (T_SZ * K_TAGS * 4) / (16 * N_CH); ++it) { // 24
            const unsigned off = (unsigned)(it * N_CH + tid) * 16u;
            asm volatile("global_load_async_to_lds_b128 %0, %1, %2"
                         :: "v"(ldsBase + off), "v"(off), "s"(gRow)
                         : "memory");
        }
        asm volatile("s_wait_asynccnt 0" ::: "memory");
    }
    __syncthreads();

    // exp(trans) matrix (constant per problem)
    float w[3][3];
#pragma unroll
    for (int k = 0; k < 9; ++k) w[k / 3][k % 3] = __builtin_exp2f(sTrans[k] * L2E);

    const int c  = tid;
    const int s0 = c * L_CH;
    const int* tagRow = tags + (size_t)b * T_SZ;

    // p = identity (exp-domain identity of the semiring); per-row 2^cnt scale
    float p[3][3] = {{1.f,0.f,0.f},{0.f,1.f,0.f},{0.f,0.f,1.f}};
    int   cnt[3]  = {0, 0, 0};
    float emit_s = 0.f, trans_s = 0.f;
    int prevTag = (s0 > 0) ? tagRow[s0 - 1] : 0;

#pragma unroll 4
    for (int j = 0; j < L_CH; ++j) {
        const int t = s0 + j;
        const float e0 = sLog[3 * t + 0];
        const float e1 = sLog[3 * t + 1];
        const float e2 = sLog[3 * t + 2];
        const int tg = tagRow[t];
        emit_s += (tg == 0) ? e0 : ((tg == 1) ? e1 : e2);
        if (t > 0) {                       // step matrices exist for t >= 1
            trans_s += sTrans[prevTag * 3 + tg];
            const float x0 = __builtin_exp2f(e0 * L2E);
            const float x1 = __builtin_exp2f(e1 * L2E);
            const float x2 = __builtin_exp2f(e2 * L2E);
            float q[3][3];
#pragma unroll
            for (int i = 0; i < 3; ++i) {
                q[i][0] = (p[i][0]*w[0][0] + p[i][1]*w[1][0] + p[i][2]*w[2][0]) * x0;
                q[i][1] = (p[i][0]*w[0][1] + p[i][1]*w[1][1] + p[i][2]*w[2][1]) * x1;
                q[i][2] = (p[i][0]*w[0][2] + p[i][1]*w[1][2] + p[i][2]*w[2][2]) * x2;
            }
#pragma unroll
            for (int i = 0; i < 3; ++i) {
                p[i][0] = q[i][0]; p[i][1] = q[i][1]; p[i][2] = q[i][2];
            }
            if ((j & 3) == 3) {            // exact power-of-two rescale
#pragma unroll
                for (int i = 0; i < 3; ++i) {
                    const float m  = fmaxf(p[i][0], fmaxf(p[i][1], p[i][2]));
                    const int   ex = (int)((__float_as_uint(m) >> 23) & 0xffu) - 127;
                    const float sc = __uint_as_float((unsigned)(127 - ex) << 23);
                    p[i][0] *= sc; p[i][1] *= sc; p[i][2] *= sc;
                    cnt[i] += ex;
                }
            }
        }
        prevTag = tg;
    }

    // chunk matrix in log domain: L[i][j] = (log2 p + cnt_i) * ln2
    float* dst = chunkM + ((size_t)b * N_CH + c) * 9;
#pragma unroll
    for (int i = 0; i < 3; ++i)
#pragma unroll
        for (int jj = 0; jj < 3; ++jj)
            dst[i * 3 + jj] = (__builtin_log2f(p[i][jj]) + (float)cnt[i]) * LN2;

    // block-reduce emit+trans partials (fixed-order tree: deterministic)
    __syncthreads();
    sRed[tid] = emit_s + trans_s;
    __syncthreads();
#pragma unroll
    for (int st = N_CH / 2; st > 0; st >>= 1) {
        if (tid < st) sRed[tid] += sRed[tid + st];
        __syncthreads();
    }
    if (tid == 0) scoreOut[b] = sRed[0];
}

// ---------------------------------------------------------------------------
// Phase 2: one block per batch. Log-semiring tree reduction of the 128 chunk
// matrices, then log_z = lse(alpha0 ∘ M_total).
// ---------------------------------------------------------------------------
__launch_bounds__(128)
__global__ void crf_phase2(const float* __restrict__ logits,
                           const float* __restrict__ chunkM,
                           float* __restrict__ logzOut) // [B]
{
    __shared__ float sM[N_CH * 9];
    const int b   = blockIdx.x;
    const int tid = threadIdx.x;

    const float* src = chunkM + (size_t)b * N_CH * 9;
    for (int i = tid; i < N_CH * 9; i += 128) sM[i] = src[i];
    __syncthreads();

    for (int s = 1; s < N_CH; s <<= 1) {
        const int pairs = N_CH / (2 * s);
        if (tid < pairs) {
            float*       A  = &sM[(tid * 2 * s) * 9];       // earlier chunk
            const float* Bm = &sM[(tid * 2 * s + s) * 9];   // later chunk
            float C[9];
#pragma unroll
            for (int i = 0; i < 3; ++i)
#pragma unroll
                for (int jj = 0; jj < 3; ++jj)
                    C[i * 3 + jj] = lse3(A[i * 3 + 0] + Bm[0 * 3 + jj],
                                         A[i * 3 + 1] + Bm[1 * 3 + jj],
                                         A[i * 3 + 2] + Bm[2 * 3 + jj]);
#pragma unroll
            for (int k = 0; k < 9; ++k) A[k] = C[k];
        }
        __syncthreads();
    }

    if (tid == 0) {
        const float* l0 = logits + (size_t)b * (T_SZ * K_TAGS);
        const float a0 = l0[0], a1 = l0[1], a2 = l0[2];   // alpha0 = emit[0]
        float al[3];
#pragma unroll
        for (int jj = 0; jj < 3; ++jj)
            al[jj] = lse3(a0 + sM[0 * 3 + jj],
                          a1 + sM[1 * 3 + jj],
                          a2 + sM[2 * 3 + jj]);
        logzOut[b] = lse3(al[0], al[1], al[2]);
    }
}

// ---------------------------------------------------------------------------
// Phase 3: deterministic final reduction: out = sum_b (log_z[b] - score[b]).
// ---------------------------------------------------------------------------
__launch_bounds__(256)
__global__ void crf_phase3(const float* __restrict__ score,
                           const float* __restrict__ logz,
                           float* __restrict__ out)
{
    __shared__ float sr[256];
    const int tid = threadIdx.x;
    float acc = 0.f;
    for (int i = tid; i < B_SZ; i += 256) acc += logz[i] - score[i];
    sr[tid] = acc;
    __syncthreads();
#pragma unroll
    for (int st = 128; st > 0; st >>= 1) {
        if (tid < st) sr[tid] += sr[tid + st];
        __syncthreads();
    }
    if (tid == 0) out[0] = sr[0];
}

extern "C" void kernel_launch(void* const* d_in, const int* in_sizes, int n_in,
                              void* d_out, int out_size, void* d_ws, size_t ws_size,
                              hipStream_t stream) {
    const float* logits = (const float*)d_in[0];   // (2048,4096,3) f32
    const float* trans  = (const float*)d_in[1];   // (3,3) f32
    const int*   tags   = (const int*)d_in[2];     // (2048,4096) int

    float* out    = (float*)d_out;
    // workspace layout: chunk matrices (9.44 MB) | score[B] | logz[B]
    float* chunkM = (float*)d_ws;
    float* score  = chunkM + (size_t)B_SZ * N_CH * 9;
    float* logz   = score + B_SZ;

    crf_phase1<<<B_SZ, N_CH, 0, stream>>>(logits, trans, tags, chunkM, score);
    crf_phase2<<<B_SZ, 128, 0, stream>>>(logits, chunkM, logz);
    crf_phase3<<<1, 256, 0, stream>>>(score, logz, out);
}